// LearnableLaplacianBasis_7146825581000
// MI455X (gfx1250) — compile-verified
//
#include <hip/hip_runtime.h>
#include <math.h>

// ---------------------------------------------------------------------------
// LearnableLaplacianBasis for MI455X (gfx1250, wave32)
//   B=32, T=2048, D=512, K=128
//   Stage 1: spectral-norm sigma (1 workgroup, LDS reductions)
//   Stage 2: u = x @ (W/sigma)^T + b  via V_WMMA_F32_16X16X4_F32.
//            W staged in LDS (64-deep chunks, padded rows, conflict-free
//            ds_load_b64), software-pipelined operand double buffering so
//            loads batch instead of serializing against each WMMA.
//            u staged into C-half of d_out (no big workspace needed).
//   Stage 3: chunked linear-recurrence scan (3 passes) for B*K*C parallelism.
// ---------------------------------------------------------------------------

typedef float v2f __attribute__((ext_vector_type(2)));
typedef float v8f __attribute__((ext_vector_type(8)));

constexpr int Bc = 32;
constexpr int Tc = 2048;
constexpr int Dc = 512;
constexpr int Kc = 128;
constexpr int Cc = 16;          // scan chunks
constexpr int Lc = Tc / Cc;     // chunk length = 128
constexpr int OUTW = 2 * Kc;    // 256 floats per (b,t) row of output

constexpr int DCH  = 64;        // GEMM depth chunk staged in LDS
constexpr int NQ   = DCH / 4;   // 16 wmma depth-steps per chunk
constexpr int LSTR = DCH + 4;   // padded LDS row stride (68): bank stride 4,
                                // 16B-aligned rows (272 B), conflict-free

__device__ __forceinline__ float softplusf(float x) {
    // matches jax.nn.softplus = logaddexp(x, 0)
    return fmaxf(x, 0.0f) + log1pf(expf(-fabsf(x)));
}

// Compute a_t = rho * e^{i theta} for one (b,t,k) element.
__device__ __forceinline__ void pole_step(float a0, float o0,
                                          float am, float om,
                                          float tm, float dtv,
                                          float& ar, float& ai) {
    float scale = expf(tm);
    float alpha = a0 * expf(am) * scale;
    float omega = o0 * expf(om) * scale;
    float rho   = expf(-alpha * dtv);
    float theta = omega * dtv;
    float st, ct;
    sincosf(theta, &st, &ct);
    ar = rho * ct;
    ai = rho * st;
}

// ---------------------------------------------------------------------------
// Kernel 1: spectral norm power iteration -> ws_sigma[0] = 1/sigma
// ---------------------------------------------------------------------------
__global__ void sn_sigma_kernel(const float* __restrict__ W,
                                const float* __restrict__ u,
                                float* __restrict__ ws_sigma) {
    __shared__ float v[Dc];
    __shared__ float red[128];
    const int tid = threadIdx.x;

    // v = W^T u  (each thread owns 4 d-slots)
    float ss = 0.0f;
    for (int d = tid; d < Dc; d += 128) {
        float acc = 0.0f;
        for (int k = 0; k < Kc; ++k)
            acc = fmaf(W[(size_t)k * Dc + d], u[k], acc);
        v[d] = acc;
        ss += acc * acc;
    }
    red[tid] = ss;
    __syncthreads();
    for (int s = 64; s > 0; s >>= 1) {
        if (tid < s) red[tid] += red[tid + s];
        __syncthreads();
    }
    const float inv_nv = 1.0f / (sqrtf(red[0]) + 1e-6f);
    __syncthreads();                       // everyone has read red[0]
    for (int d = tid; d < Dc; d += 128) v[d] *= inv_nv;
    __syncthreads();

    // wv = W v  (thread tid owns row k=tid)
    float acc = 0.0f;
    {
        const int k = tid;
        const float* wr = W + (size_t)k * Dc;
        for (int d = 0; d < Dc; ++d)
            acc = fmaf(wr[d], v[d], acc);
    }
    red[tid] = acc * acc;
    __syncthreads();
    for (int s = 64; s > 0; s >>= 1) {
        if (tid < s) red[tid] += red[tid + s];
        __syncthreads();
    }
    if (tid == 0) {
        float dot = red[0];                // = wv . wv
        float nrm = sqrtf(dot);
        // sigma = dot/(nrm+eps)  ->  inv_sigma = (nrm+eps)/dot
        ws_sigma[0] = (nrm + 1e-6f) / dot;
    }
}

// ---------------------------------------------------------------------------
// Kernel 2: u = x @ (W/sigma)^T + b   via V_WMMA_F32_16X16X4_F32
//   blockDim = 256 (8 waves); each wave owns a 16-row slab of M=B*T and all
//   8 n-tiles of K=128. W staged via LDS in 64-deep chunks; operands double
//   buffered so loads pipeline against WMMA. u written to out[m*256 + n].
// ---------------------------------------------------------------------------
__global__ void proj_gemm_kernel(const float* __restrict__ x,
                                 const float* __restrict__ W,
                                 const float* __restrict__ bias,
                                 const float* __restrict__ ws_sigma,
                                 float* __restrict__ out) {
    __shared__ float Wlds[Kc * LSTR];             // 128 x 68 floats = 34 KB

    const int tid    = threadIdx.x;
    const int wave   = tid >> 5;
    const int lane   = tid & 31;
    const int lane15 = lane & 15;
    const int khalf  = (lane >> 4) << 1;          // 0 or 2 (K-slot within 4)
    const int m0     = (blockIdx.x * 8 + wave) * 16;

    const v8f vzero = {0.f, 0.f, 0.f, 0.f, 0.f, 0.f, 0.f, 0.f};
    v8f acc[8];
#pragma unroll
    for (int nt = 0; nt < 8; ++nt) acc[nt] = vzero;

    // A operand: row M = m0 + lane15, depth slots khalf..khalf+1
    const float* arow = x + (size_t)(m0 + lane15) * Dc + khalf;
    // B operand base within LDS: row n = nt*16 + lane15
    const int brow = lane15 * LSTR + khalf;

    for (int c = 0; c < Dc / DCH; ++c) {
        const int d_base = c * DCH;

        __syncthreads();   // all consumers of previous chunk done
        // Cooperative fill: 128 rows x 16 float4, coalesced, uniform trip
        for (int i = tid; i < Kc * NQ; i += 256) {
            const int n  = i >> 4;                // 16 float4 per row
            const int dq = i & 15;
            *(float4*)(&Wlds[n * LSTR + dq * 4]) =
                *(const float4*)(W + (size_t)n * Dc + d_base + dq * 4);
        }
        __syncthreads();

        // prime the operand pipeline
        v2f a_cur = *(const v2f*)(arow + d_base);
        v2f b_cur[8];
#pragma unroll
        for (int nt = 0; nt < 8; ++nt)
            b_cur[nt] = *(const v2f*)(&Wlds[brow + nt * 16 * LSTR]);

#pragma unroll 4
        for (int d4 = 0; d4 < NQ; ++d4) {
            const int dn = (d4 + 1) & (NQ - 1);   // wraps on last (discarded)
            v2f a_nxt = *(const v2f*)(arow + d_base + dn * 4);
            v2f b_nxt[8];
#pragma unroll
            for (int nt = 0; nt < 8; ++nt)
                b_nxt[nt] = *(const v2f*)(&Wlds[brow + nt * 16 * LSTR + dn * 4]);
#pragma unroll
            for (int nt = 0; nt < 8; ++nt)
                acc[nt] = __builtin_amdgcn_wmma_f32_16x16x4_f32(
                    /*neg_a=*/false, a_cur, /*neg_b=*/false, b_cur[nt],
                    /*c_mod=*/(short)0, acc[nt],
                    /*reuse_a=*/false, /*reuse_b=*/false);
            a_cur = a_nxt;
#pragma unroll
            for (int nt = 0; nt < 8; ++nt) b_cur[nt] = b_nxt[nt];
        }
    }

    const float inv_s = ws_sigma[0];
    const int mrow = m0 + ((lane >> 4) << 3);     // +8 for upper half-wave
#pragma unroll
    for (int nt = 0; nt < 8; ++nt) {
        const int n = nt * 16 + lane15;
        const float bc = bias[n];
#pragma unroll
        for (int i = 0; i < 8; ++i) {
            out[(size_t)(mrow + i) * OUTW + n] = acc[nt][i] * inv_s + bc;
        }
    }
}

// ---------------------------------------------------------------------------
// Kernel 3a: local scan per chunk; z local (z0=0) overwrites u in d_out,
//            per-chunk complex product -> Pws.   grid (Cc, Bc), block Kc.
// ---------------------------------------------------------------------------
__global__ void scan1_kernel(const float* __restrict__ alpha_mod,
                             const float* __restrict__ omega_mod,
                             const float* __restrict__ tau_mod,
                             const float* __restrict__ dt,
                             const float* __restrict__ s_real,
                             const float* __restrict__ s_imag,
                             const float* __restrict__ tau_raw,
                             float* __restrict__ out,
                             float2* __restrict__ Pws) {
    const int k = threadIdx.x;
    const int j = blockIdx.x;
    const int b = blockIdx.y;

    const float tau = softplusf(tau_raw[0]) + 1e-3f;
    const float a0  = (softplusf(s_real[k]) + 1e-6f) * tau;
    const float o0  = s_imag[k] * tau;

    float zr = 0.f, zi = 0.f;
    float Pr = 1.f, Pi = 0.f;
    const int t0 = j * Lc;

#pragma unroll 2
    for (int tt = 0; tt < Lc; ++tt) {
        const int t = t0 + tt;
        const size_t bt  = (size_t)b * Tc + t;
        const size_t idx = bt * Kc + k;
        if (tt + 16 < Lc) {
            __builtin_prefetch(&alpha_mod[idx + 16 * Kc], 0, 0);
            __builtin_prefetch(&omega_mod[idx + 16 * Kc], 0, 0);
        }
        float ar, ai;
        pole_step(a0, o0, alpha_mod[idx], omega_mod[idx],
                  tau_mod[bt], dt[bt], ar, ai);

        const size_t o = bt * OUTW + k;
        const float uu = out[o];                       // staged u
        const float nzr = fmaf(ar, zr, fmaf(-ai, zi, uu));
        const float nzi = fmaf(ar, zi, ai * zr);
        zr = nzr; zi = nzi;
        out[o]      = zr;                              // local C
        out[o + Kc] = zi;                              // local S

        const float nPr = ar * Pr - ai * Pi;
        const float nPi = ar * Pi + ai * Pr;
        Pr = nPr; Pi = nPi;
    }
    Pws[((size_t)b * Cc + j) * Kc + k] = make_float2(Pr, Pi);
}

// ---------------------------------------------------------------------------
// Kernel 3b: sequential carry combine across chunks (tiny).
//   carry_{j+1} = local_end_j + P_j * carry_j
// ---------------------------------------------------------------------------
__global__ void scan2_kernel(const float* __restrict__ out,
                             const float2* __restrict__ Pws,
                             float2* __restrict__ Cws) {
    const int k = threadIdx.x;
    const int b = blockIdx.x;
    float cr = 0.f, ci = 0.f;
    for (int j = 0; j < Cc; ++j) {
        Cws[((size_t)b * Cc + j) * Kc + k] = make_float2(cr, ci);
        if (j < Cc - 1) {
            const size_t bt = (size_t)b * Tc + (size_t)(j + 1) * Lc - 1;
            const size_t o  = bt * OUTW + k;
            const float er = out[o];
            const float ei = out[o + Kc];
            const float2 P = Pws[((size_t)b * Cc + j) * Kc + k];
            const float ncr = er + P.x * cr - P.y * ci;
            const float nci = ei + P.x * ci + P.y * cr;
            cr = ncr; ci = nci;
        }
    }
}

// ---------------------------------------------------------------------------
// Kernel 3c: apply carries: z_t += prefix_t * carry_j (recompute prefixes,
//            bit-identical pole_step as pass 1).
// ---------------------------------------------------------------------------
__global__ void scan3_kernel(const float* __restrict__ alpha_mod,
                             const float* __restrict__ omega_mod,
                             const float* __restrict__ tau_mod,
                             const float* __restrict__ dt,
                             const float* __restrict__ s_real,
                             const float* __restrict__ s_imag,
                             const float* __restrict__ tau_raw,
                             const float2* __restrict__ Cws,
                             float* __restrict__ out) {
    const int k = threadIdx.x;
    const int j = blockIdx.x;
    const int b = blockIdx.y;
    if (j == 0) return;                         // carry_0 == 0 exactly

    const float2 c = Cws[((size_t)b * Cc + j) * Kc + k];
    const float tau = softplusf(tau_raw[0]) + 1e-3f;
    const float a0  = (softplusf(s_real[k]) + 1e-6f) * tau;
    const float o0  = s_imag[k] * tau;

    float pr = 1.f, pi = 0.f;
    const int t0 = j * Lc;
#pragma unroll 2
    for (int tt = 0; tt < Lc; ++tt) {
        const int t = t0 + tt;
        const size_t bt  = (size_t)b * Tc + t;
        const size_t idx = bt * Kc + k;
        float ar, ai;
        pole_step(a0, o0, alpha_mod[idx], omega_mod[idx],
                  tau_mod[bt], dt[bt], ar, ai);
        const float npr = ar * pr - ai * pi;
        const float npi = ar * pi + ai * pr;
        pr = npr; pi = npi;

        const size_t o = bt * OUTW + k;
        out[o]      += pr * c.x - pi * c.y;
        out[o + Kc] += pr * c.y + pi * c.x;
    }
}

// ---------------------------------------------------------------------------
extern "C" void kernel_launch(void* const* d_in, const int* in_sizes, int n_in,
                              void* d_out, int out_size, void* d_ws, size_t ws_size,
                              hipStream_t stream) {
    (void)in_sizes; (void)n_in; (void)out_size; (void)ws_size;

    const float* x         = (const float*)d_in[0];   // [B,T,D]
    const float* dt        = (const float*)d_in[1];   // [B,T]
    const float* alpha_mod = (const float*)d_in[2];   // [B,T,K]
    const float* omega_mod = (const float*)d_in[3];   // [B,T,K]
    const float* tau_mod   = (const float*)d_in[4];   // [B,T,1]
    const float* s_real    = (const float*)d_in[5];   // [K]
    const float* s_imag    = (const float*)d_in[6];   // [K]
    const float* tau_raw   = (const float*)d_in[7];   // [1]
    const float* W         = (const float*)d_in[8];   // [K,D]
    const float* bias      = (const float*)d_in[9];   // [K]
    const float* u_sn      = (const float*)d_in[10];  // [K]
    float* out = (float*)d_out;                       // [B,T,2K]

    // workspace layout: [0..3] inv_sigma, then chunk products, then carries
    float*  sig = (float*)d_ws;
    float2* Pws = (float2*)((char*)d_ws + 256);
    float2* Cws = (float2*)((char*)d_ws + 256 + sizeof(float2) * Bc * Cc * Kc);

    sn_sigma_kernel<<<1, 128, 0, stream>>>(W, u_sn, sig);

    const int gemm_blocks = (Bc * Tc) / 128;          // 8 waves x 16 rows
    proj_gemm_kernel<<<gemm_blocks, 256, 0, stream>>>(x, W, bias, sig, out);

    scan1_kernel<<<dim3(Cc, Bc), Kc, 0, stream>>>(alpha_mod, omega_mod, tau_mod,
                                                  dt, s_real, s_imag, tau_raw,
                                                  out, Pws);
    scan2_kernel<<<Bc, Kc, 0, stream>>>(out, Pws, Cws);
    scan3_kernel<<<dim3(Cc, Bc), Kc, 0, stream>>>(alpha_mod, omega_mod, tau_mod,
                                                  dt, s_real, s_imag, tau_raw,
                                                  Cws, out);
}